// SpanRepresentation_63840393888081
// MI455X (gfx1250) — compile-verified
//
#include <hip/hip_runtime.h>
#include <hip/hip_bf16.h>
#include <stdint.h>

// Span representation gather/concat for MI455X (gfx1250).
// Pure data movement (0 FLOPs): ~106MB streamed writes -> bandwidth bound
// (~4.8us floor @ 23.3TB/s). Strategy:
//   * CDNA5 async data mover: global_load_async_to_lds_b128 (+ s_wait_asynccnt)
//     to stage the two x rows + embedding row per span into LDS.
//   * 128-bit non-temporal stores for the write-once 106MB output stream
//     (keeps L2 available for the 6MB x tensor that is reused ~16x).
//   * wave32 / 8-wave blocks; 4KB fully-coalesced bursts per loop trip.

typedef float v4f __attribute__((ext_vector_type(4)));
// Builtin parameter types (from hipcc diagnostic): gcc-style int vector ptrs.
typedef int v4i_g __attribute__((vector_size(16)));
typedef __attribute__((address_space(1))) v4i_g* g4i_t;   // global src
typedef __attribute__((address_space(3))) v4i_g* l4i_t;   // LDS dst

#define NB     16      // batch
#define NL     128     // seq len (batch_max_seq_len in setup)
#define ND     768     // hidden
#define NW     128     // width-embedding dim
#define NSPANS 996     // sum_{w=1..8} (129 - w)
#define CROW   1664    // 2*ND + NW floats per span
#define D4     192     // ND  / 4
#define W4     32      // NW  / 4
#define QROW   416     // CROW/ 4 float4 per span row
#define TPB    256     // 8 waves (wave32)

#if __has_builtin(__builtin_amdgcn_global_load_async_to_lds_b128) && \
    __has_builtin(__builtin_amdgcn_s_wait_asynccnt)
#define USE_ASYNC_LDS 1
#else
#define USE_ASYNC_LDS 0
#endif

__device__ __forceinline__ void span_decode(int s, int& w, int& start) {
  // spans are ordered width-major: w=1 occupies [0,128), w=2 [128,255), ...
  int ww = 1;
  int base = 0;
  if (s >= 128) { ww = 2; base = 128; }
  if (s >= 255) { ww = 3; base = 255; }
  if (s >= 381) { ww = 4; base = 381; }
  if (s >= 506) { ww = 5; base = 506; }
  if (s >= 630) { ww = 6; base = 630; }
  if (s >= 753) { ww = 7; base = 753; }
  if (s >= 875) { ww = 8; base = 875; }
  w = ww;
  start = s - base;
}

__global__ __launch_bounds__(TPB)
void span_rep_kernel(const float* __restrict__ x,
                     const float* __restrict__ wemb,
                     float* __restrict__ out)
{
  const int s = blockIdx.x;   // span id    [0, NSPANS)
  const int b = blockIdx.y;   // batch id   [0, NB)

  int w, start;
  span_decode(s, w, start);
  const int end    = start + w - 1;
  // searchsorted(BUCKET_BINS, w, 'right')-1 for w in 1..8 -> {1,2,3,4,5,5,6,7}
  const int bucket = (w <= 5) ? w : (w - 1);

  const v4f* __restrict__ rowS = (const v4f*)(x + ((size_t)b * NL + start) * ND);
  const v4f* __restrict__ rowE = (const v4f*)(x + ((size_t)b * NL + end)   * ND);
  const v4f* __restrict__ rowW = (const v4f*)(wemb + (size_t)bucket * NW);
  v4f* __restrict__ orow = (v4f*)(out + ((size_t)b * NSPANS + s) * CROW);

#if USE_ASYNC_LDS
  __shared__ v4f lds[QROW];   // 6656 B staging tile

  for (int q = threadIdx.x; q < QROW; q += TPB) {
    const v4f* src = (q < D4)     ? (rowS + q)
                   : (q < 2 * D4) ? (rowE + (q - D4))
                                  : (rowW + (q - 2 * D4));
    __builtin_amdgcn_global_load_async_to_lds_b128(
        (g4i_t)(uintptr_t)src,
        (l4i_t)(uint32_t)(uintptr_t)&lds[q],
        0, 0);
  }
  __builtin_amdgcn_s_wait_asynccnt(0);   // ASYNCcnt == 0: LDS tile resident
  __syncthreads();                       // make it visible across all 8 waves

  for (int q = threadIdx.x; q < QROW; q += TPB) {
    __builtin_nontemporal_store(lds[q], orow + q);
  }
#else
  for (int q = threadIdx.x; q < QROW; q += TPB) {
    const v4f v = (q < D4)     ? rowS[q]
                : (q < 2 * D4) ? rowE[q - D4]
                               : rowW[q - 2 * D4];
    __builtin_nontemporal_store(v, orow + q);
  }
#endif

  // Second output: span_indices (n_spans, 2), exact small ints as f32.
  if (b == 0 && threadIdx.x == 0) {
    float* oidx = out + (size_t)NB * NSPANS * CROW + (size_t)s * 2;
    oidx[0] = (float)start;
    oidx[1] = (float)end;
  }
}

extern "C" void kernel_launch(void* const* d_in, const int* in_sizes, int n_in,
                              void* d_out, int out_size, void* d_ws, size_t ws_size,
                              hipStream_t stream) {
  (void)in_sizes; (void)n_in; (void)out_size; (void)d_ws; (void)ws_size;
  const float* x    = (const float*)d_in[0];  // (16, 128, 768) f32
  const float* wemb = (const float*)d_in[1];  // (14, 128)      f32
  // d_in[2] = batch_max_seq_len scalar (==128); geometry is compile-time here.
  float* out = (float*)d_out;

  dim3 grid(NSPANS, NB);
  span_rep_kernel<<<grid, TPB, 0, stream>>>(x, wemb, out);
}